// GCNModel_30554397343903
// MI455X (gfx1250) — compile-verified
//
#include <hip/hip_runtime.h>
#include <hip/hip_bf16.h>

#define NN 50000
#define NE 600000
#define DIN 128
#define DH 128
#define DOUT 64

typedef __attribute__((ext_vector_type(2))) float v2f;
typedef __attribute__((ext_vector_type(8))) float v8f;
typedef __attribute__((ext_vector_type(4))) int v4i;

__device__ __forceinline__ v8f wmma_f32(v2f a, v2f b, v8f c) {
    // (neg_a, A, neg_b, B, c_mod, C, reuse_a, reuse_b)
    return __builtin_amdgcn_wmma_f32_16x16x4_f32(false, a, false, b, (short)0, c,
                                                 false, false);
}

// ---- LDS staging helper: async global->LDS when available, else plain copy ----
#if __has_builtin(__builtin_amdgcn_global_load_async_to_lds_b128) && \
    __has_builtin(__builtin_amdgcn_s_wait_asynccnt)
#define HAVE_ASYNC_LDS 1
#else
#define HAVE_ASYNC_LDS 0
#endif

__device__ __forceinline__ void stage_to_lds(float* lds_dst, const float* gsrc, int nFloats,
                                             int tid, int nthreads) {
#if HAVE_ASYNC_LDS
    typedef __attribute__((address_space(1))) v4i* gv4_t;   // global int4*
    typedef __attribute__((address_space(3))) v4i* lv4_t;   // LDS int4*
    for (int i = tid * 4; i < nFloats; i += nthreads * 4) {
        gv4_t gp = (gv4_t)(uintptr_t)(gsrc + i);     // AS1 == generic global bits
        lv4_t lp = (lv4_t)(uintptr_t)(lds_dst + i);  // low 32 bits == LDS offset
        __builtin_amdgcn_global_load_async_to_lds_b128(gp, lp, 0, 0);
    }
    __builtin_amdgcn_s_wait_asynccnt(0);
#else
    for (int i = tid * 4; i < nFloats; i += nthreads * 4)
        *(float4*)(lds_dst + i) = *(const float4*)(gsrc + i);
#endif
    __syncthreads();
}

// ---------------- degree / normalization ----------------
__global__ void k_fill_ones(float* p, int n) {
    int i = blockIdx.x * blockDim.x + threadIdx.x;
    if (i < n) p[i] = 1.0f;
}

__global__ void k_deg_accum(const int* __restrict__ dst, float* __restrict__ deg, int e) {
    int i = blockIdx.x * blockDim.x + threadIdx.x;
    if (i < e) unsafeAtomicAdd(&deg[dst[i]], 1.0f);
}

__global__ void k_to_dinv(float* p, int n) {
    int i = blockIdx.x * blockDim.x + threadIdx.x;
    if (i < n) p[i] = rsqrtf(p[i]);   // deg >= 1 (self-loop)
}

// ---------------- fp32 WMMA GEMM: C[M,N] = A[M,K] @ B[K,N] ----------------
// B staged in LDS once per block; each wave computes MT m-tiles x 1 n-tile,
// reusing one B fragment across MT v_wmma_f32_16x16x4_f32 per k-step.
template <int K, int NCOLS, int MT>
__global__ void k_gemm_wmma(const float* __restrict__ A, const float* __restrict__ B,
                            float* __restrict__ C, int mGroups) {
    constexpr int nTiles = NCOLS / 16;
    __shared__ __align__(16) float sB[K * NCOLS];
    stage_to_lds(sB, B, K * NCOLS, threadIdx.x, blockDim.x);

    int wave   = blockIdx.x * (blockDim.x >> 5) + (threadIdx.x >> 5);
    int mGroup = wave / nTiles;
    int nTile  = wave % nTiles;
    if (mGroup >= mGroups) return;            // wave-uniform (after barrier)

    int lane = threadIdx.x & 31;
    int m    = lane & 15;
    int koff = (lane >> 4) << 1;              // lanes 0-15: K=k,k+1 ; 16-31: K=k+2,k+3

    const float* arow[MT];
#pragma unroll
    for (int j = 0; j < MT; ++j)
        arow[j] = A + (size_t)((mGroup * MT + j) * 16 + m) * K;
    const float* bcol = sB + nTile * 16 + m;

    v8f acc[MT] = {};
    for (int k = 0; k < K; k += 4) {
        v2f b;
        b.x = bcol[(k + koff) * NCOLS];
        b.y = bcol[(k + koff + 1) * NCOLS];
#pragma unroll
        for (int j = 0; j < MT; ++j) {
            v2f a = *(const v2f*)(arow[j] + k + koff);
            acc[j] = wmma_f32(a, b, acc[j]);
        }
    }

    // C layout: VGPR r -> (M = r + 8*(lane>=16), N = lane&15)
    int col = nTile * 16 + (lane & 15);
#pragma unroll
    for (int j = 0; j < MT; ++j) {
        int rowBase = (mGroup * MT + j) * 16 + (lane >> 4) * 8;
#pragma unroll
        for (int r = 0; r < 8; ++r)
            C[(size_t)(rowBase + r) * NCOLS + col] = acc[j][r];
    }
}

// ---------------- GCN aggregation ----------------
template <int D>
__global__ void k_self_scale(const float* __restrict__ XW, const float* __restrict__ dinv,
                             float* __restrict__ agg, int nD) {
    int i = blockIdx.x * blockDim.x + threadIdx.x;
    if (i >= nD) return;
    float di = dinv[i / D];
    agg[i] = XW[i] * di * di;                 // self-loop term
}

template <int D>
__global__ void k_edge_scatter(const float* __restrict__ XW, const int* __restrict__ src,
                               const int* __restrict__ dst, const float* __restrict__ dinv,
                               float* __restrict__ agg, int eD) {
    int i = blockIdx.x * blockDim.x + threadIdx.x;
    if (i >= eD) return;
    int e = i / D;
    int f = i % D;
    int s = src[e], d = dst[e];
    float nrm = dinv[s] * dinv[d];
    unsafeAtomicAdd(&agg[(size_t)d * D + f], XW[(size_t)s * D + f] * nrm);
}

template <int D, bool RELU>
__global__ void k_bias_act(float* __restrict__ agg, const float* __restrict__ b, int nD) {
    int i = blockIdx.x * blockDim.x + threadIdx.x;
    if (i >= nD) return;
    float v = agg[i] + b[i % D];
    if (RELU) v = v > 0.0f ? v : 0.0f;
    agg[i] = v;
}

// ---------------- fused edge decoder ----------------
// Wm1 (128x64 = 32KB) staged in LDS per block. One wave handles 16 edges:
// A = [H[src] | H[dst]] (16x128 gathered), four 16x16 WMMA C tiles cover N=64,
// then ReLU + 64->1 projection reduced with shfl_xor inside 16-lane halves.
__global__ void k_decoder_wmma(const float* __restrict__ H,   // [NN, 64]
                               const int* __restrict__ src, const int* __restrict__ dst,
                               const float* __restrict__ Wm1, // [128, 64]
                               const float* __restrict__ bm1, // [64]
                               const float* __restrict__ Wm2, // [64]
                               const float* __restrict__ bm2, // [1]
                               float* __restrict__ out, int eTiles) {
    __shared__ __align__(16) float sW[128 * 64];
    stage_to_lds(sW, Wm1, 128 * 64, threadIdx.x, blockDim.x);

    int wave = blockIdx.x * (blockDim.x >> 5) + (threadIdx.x >> 5);
    if (wave >= eTiles) return;               // wave-uniform (after barrier)

    int lane = threadIdx.x & 31;
    int m    = lane & 15;
    int half = lane >> 4;
    int koff = half << 1;

    int e = wave * 16 + m;
    const float* hs = H + (size_t)src[e] * DOUT;
    const float* hd = H + (size_t)dst[e] * DOUT;

    v8f acc[4] = {};
    for (int k = 0; k < 128; k += 4) {
        int kk = k + koff;
        const float* ap = (kk < 64) ? (hs + kk) : (hd + (kk - 64));
        v2f a = *(const v2f*)ap;
        const float* w0 = sW + kk * 64 + m;
#pragma unroll
        for (int t = 0; t < 4; ++t) {
            v2f b;
            b.x = w0[t * 16];
            b.y = w0[64 + t * 16];
            acc[t] = wmma_f32(a, b, acc[t]);
        }
    }

    // h = relu(acc + bm1[n]); partial[r] = sum_n h * Wm2[n] over this lane's n's
    float partial[8];
#pragma unroll
    for (int r = 0; r < 8; ++r) partial[r] = 0.0f;
#pragma unroll
    for (int t = 0; t < 4; ++t) {
        int n   = t * 16 + m;
        float w = Wm2[n];
        float b = bm1[n];
#pragma unroll
        for (int r = 0; r < 8; ++r) {
            float h = acc[t][r] + b;
            h = h > 0.0f ? h : 0.0f;
            partial[r] += h * w;
        }
    }
#pragma unroll
    for (int off = 8; off >= 1; off >>= 1) {
#pragma unroll
        for (int r = 0; r < 8; ++r)
            partial[r] += __shfl_xor(partial[r], off, 32);
    }
    if (m == 0) {
        float b2 = bm2[0];
        int rowBase = wave * 16 + half * 8;   // lane0 -> M 0..7, lane16 -> M 8..15
#pragma unroll
        for (int r = 0; r < 8; ++r)
            out[rowBase + r] = partial[r] + b2;
    }
}

extern "C" void kernel_launch(void* const* d_in, const int* in_sizes, int n_in,
                              void* d_out, int out_size, void* d_ws, size_t ws_size,
                              hipStream_t stream) {
    const float* X   = (const float*)d_in[0];
    const int*   edg = (const int*)d_in[1];
    const float* Wg1 = (const float*)d_in[2];
    const float* bg1 = (const float*)d_in[3];
    const float* Wg2 = (const float*)d_in[4];
    const float* bg2 = (const float*)d_in[5];
    const float* Wm1 = (const float*)d_in[6];
    const float* bm1 = (const float*)d_in[7];
    const float* Wm2 = (const float*)d_in[8];
    const float* bm2 = (const float*)d_in[9];
    float* out = (float*)d_out;

    const int* src = edg;
    const int* dst = edg + NE;

    // workspace layout (floats): dinv | XW1 | A1(H1) | XW2 | A2(H2)  ~77 MB
    float* dinv = (float*)d_ws;
    float* XW1  = dinv + NN;
    float* A1   = XW1 + (size_t)NN * DH;
    float* XW2  = A1  + (size_t)NN * DH;
    float* A2   = XW2 + (size_t)NN * DOUT;

    const int T = 256;
    // normalization
    k_fill_ones<<<(NN + T - 1) / T, T, 0, stream>>>(dinv, NN);
    k_deg_accum<<<(NE + T - 1) / T, T, 0, stream>>>(dst, dinv, NE);
    k_to_dinv<<<(NN + T - 1) / T, T, 0, stream>>>(dinv, NN);

    // 3125 m-tiles = 625 groups of 5 (exact)
    const int mGroups = NN / (16 * 5);        // 625
    // layer 1: XW1 = X @ Wg1   (50000x128 @ 128x128)
    {
        int waves = mGroups * (DH / 16);
        k_gemm_wmma<DIN, DH, 5><<<(waves + 7) / 8, T, 0, stream>>>(X, Wg1, XW1, mGroups);
    }
    k_self_scale<DH><<<((NN * DH) + T - 1) / T, T, 0, stream>>>(XW1, dinv, A1, NN * DH);
    k_edge_scatter<DH><<<((NE * DH) + T - 1) / T, T, 0, stream>>>(XW1, src, dst, dinv, A1, NE * DH);
    k_bias_act<DH, true><<<((NN * DH) + T - 1) / T, T, 0, stream>>>(A1, bg1, NN * DH);

    // layer 2: XW2 = H1 @ Wg2  (50000x128 @ 128x64)
    {
        int waves = mGroups * (DOUT / 16);
        k_gemm_wmma<DH, DOUT, 5><<<(waves + 7) / 8, T, 0, stream>>>(A1, Wg2, XW2, mGroups);
    }
    k_self_scale<DOUT><<<((NN * DOUT) + T - 1) / T, T, 0, stream>>>(XW2, dinv, A2, NN * DOUT);
    k_edge_scatter<DOUT><<<((NE * DOUT) + T - 1) / T, T, 0, stream>>>(XW2, src, dst, dinv, A2, NE * DOUT);
    k_bias_act<DOUT, false><<<((NN * DOUT) + T - 1) / T, T, 0, stream>>>(A2, bg2, NN * DOUT);

    // decoder over 600000 edges (37500 tiles of 16)
    const int eTiles = NE / 16;
    k_decoder_wmma<<<(eTiles + 7) / 8, T, 0, stream>>>(A2, src, dst, Wm1, bm1, Wm2, bm2,
                                                       out, eTiles);
}